// Match_Decoder_58995670778419
// MI455X (gfx1250) — compile-verified
//
#include <hip/hip_runtime.h>
#include <math.h>

#define V  50257
#define H  1032
#define IN (2*H)     // 2064
#define S  8192
#define G3 (3*H)     // 3096

typedef __attribute__((ext_vector_type(2))) float v2f;
typedef __attribute__((ext_vector_type(8))) float v8f;

// ---------------- workspace layout (float offsets, all 16B aligned) --------
#define WS_X      0                      // 2064
#define WS_GX     2064                   // 3096
#define WS_GH     5160                   // 3096
#define WS_HNEW   8256                   // 1032
#define WS_V      9288                   // 1032
#define WS_VPART  10320                  // 8*1032
#define WS_SB     18576                  // 1 (+pad)
#define WS_SCORES 18580                  // 8192
#define WS_ATTNW  26772                  // 8192
#define WS_CPART  34964                  // 16*1032
#define WS_JOIN   51476                  // 2064
#define WS_LOGITS 53540                  // 50257
#define WS_RED    103800                 // 2

// ---------------- d_out layout ---------------------------------------------
#define OUT_OUTPUT 0
#define OUT_CTX    (V)
#define OUT_HID    (V + H)
#define OUT_ATTNW  (V + 2*H)

// ---------------------------------------------------------------------------
__device__ inline float waveSum(float v) {
  for (int o = 16; o; o >>= 1) v += __shfl_xor(v, o, 32);
  return v;
}
__device__ inline float waveMax(float v) {
  for (int o = 16; o; o >>= 1) v = fmaxf(v, __shfl_xor(v, o, 32));
  return v;
}

// K1: x = [ emb[word], last_context ]
__global__ void k_embed(const int* __restrict__ word, const float* __restrict__ emb,
                        const float* __restrict__ lc, float* __restrict__ x) {
  int i = blockIdx.x * blockDim.x + threadIdx.x;
  if (i >= IN) return;
  x[i] = (i < H) ? emb[(size_t)word[0] * H + i] : lc[i - H];
}

// K2: wave per row r in [0,3096): gx[r] = W_ih[r,:].x + b_ih, gh[r] = W_hh[r,:].h + b_hh
__global__ __launch_bounds__(256) void k_gates(
    const float* __restrict__ Wih, const float* __restrict__ Whh,
    const float* __restrict__ bih, const float* __restrict__ bhh,
    const float* __restrict__ x,   const float* __restrict__ h,
    float* __restrict__ gx, float* __restrict__ gh) {
  int wave = (blockIdx.x * blockDim.x + threadIdx.x) >> 5;
  int lane = threadIdx.x & 31;
  if (wave >= G3) return;
  const float4* wr = (const float4*)(Wih + (size_t)wave * IN);
  const float4* xv = (const float4*)x;
  float a = 0.f;
  for (int i = lane; i < IN/4; i += 32) {
    float4 w4 = wr[i], v4 = xv[i];
    a += w4.x*v4.x + w4.y*v4.y + w4.z*v4.z + w4.w*v4.w;
  }
  const float4* wr2 = (const float4*)(Whh + (size_t)wave * H);
  const float4* hv  = (const float4*)h;
  float b = 0.f;
  for (int i = lane; i < H/4; i += 32) {
    float4 w4 = wr2[i], v4 = hv[i];
    b += w4.x*v4.x + w4.y*v4.y + w4.z*v4.z + w4.w*v4.w;
  }
  a = waveSum(a); b = waveSum(b);
  if (lane == 0) { gx[wave] = a + bih[wave]; gh[wave] = b + bhh[wave]; }
}

// K3: GRU elementwise -> h_new (ws + d_out + joined[0:H])
__global__ void k_gru(const float* __restrict__ gx, const float* __restrict__ gh,
                      const float* __restrict__ h,  float* __restrict__ hnew,
                      float* __restrict__ joined,   float* __restrict__ hid_out) {
  int i = blockIdx.x * blockDim.x + threadIdx.x;
  if (i >= H) return;
  float r = 1.f / (1.f + expf(-(gx[i]       + gh[i])));
  float z = 1.f / (1.f + expf(-(gx[H + i]   + gh[H + i])));
  float n = tanhf(gx[2*H + i] + r * gh[2*H + i]);
  float hv = (1.f - z) * n + z * h[i];
  hnew[i] = hv; joined[i] = hv; hid_out[i] = hv;
}

// K4a: vpart[c][i] = sum_{j in chunk c} h[j] * W_a[j][i]   (column-major reduce, 8 chunks of 129)
__global__ void k_vpart(const float* __restrict__ Wa, const float* __restrict__ hn,
                        float* __restrict__ vpart) {
  int i = blockIdx.x * blockDim.x + threadIdx.x;
  if (i >= H) return;
  int j0 = blockIdx.y * 129;
  float a = 0.f;
  for (int j = j0; j < j0 + 129; ++j) a += hn[j] * Wa[(size_t)j * H + i];
  vpart[blockIdx.y * H + i] = a;
}
__global__ void k_vreduce(const float* __restrict__ vpart, float* __restrict__ v) {
  int i = blockIdx.x * blockDim.x + threadIdx.x;
  if (i >= H) return;
  float a = 0.f;
  for (int c = 0; c < 8; ++c) a += vpart[c * H + i];
  v[i] = a;
}

// K4b: sb = dot(b_a, h_new)   (single block)
__global__ __launch_bounds__(1024) void k_sb(const float* __restrict__ ba,
                                             const float* __restrict__ hn,
                                             float* __restrict__ sb) {
  __shared__ float sm[32]; __shared__ float bc;
  int tid = threadIdx.x;
  float a = 0.f;
  for (int i = tid; i < H; i += 1024) a += ba[i] * hn[i];
  a = waveSum(a);
  if ((tid & 31) == 0) sm[tid >> 5] = a;
  __syncthreads();
  if (tid < 32) { float t = sm[tid]; t = waveSum(t); if (tid == 0) bc = t; }
  __syncthreads();
  if (tid == 0) sb[0] = bc;
}

// K5: wave per row i in [0,8192): scores[i] = enc[i,:].v + sb
__global__ __launch_bounds__(256) void k_scores(
    const float* __restrict__ enc, const float* __restrict__ v,
    const float* __restrict__ sb,  float* __restrict__ scores) {
  int wave = (blockIdx.x * blockDim.x + threadIdx.x) >> 5;
  int lane = threadIdx.x & 31;
  if (wave >= S) return;
  const float4* er = (const float4*)(enc + (size_t)wave * H);
  const float4* vv = (const float4*)v;
  float a = 0.f;
  for (int i = lane; i < H/4; i += 32) {
    float4 e4 = er[i], v4 = vv[i];
    a += e4.x*v4.x + e4.y*v4.y + e4.z*v4.z + e4.w*v4.w;
  }
  a = waveSum(a);
  if (lane == 0) scores[wave] = a + sb[0];
}

// K6: softmax over 8192 (single block), writes weights to ws and d_out
__global__ __launch_bounds__(1024) void k_softmax(const float* __restrict__ scores,
                                                  float* __restrict__ w_ws,
                                                  float* __restrict__ w_out) {
  __shared__ float sm[32]; __shared__ float bc;
  int tid = threadIdx.x;
  float m = -3.4e38f;
  for (int i = tid; i < S; i += 1024) m = fmaxf(m, scores[i]);
  m = waveMax(m);
  if ((tid & 31) == 0) sm[tid >> 5] = m;
  __syncthreads();
  if (tid < 32) { float t = sm[tid]; t = waveMax(t); if (tid == 0) bc = t; }
  __syncthreads();
  float M = bc;
  __syncthreads();
  float s = 0.f;
  for (int i = tid; i < S; i += 1024) s += expf(scores[i] - M);
  s = waveSum(s);
  if ((tid & 31) == 0) sm[tid >> 5] = s;
  __syncthreads();
  if (tid < 32) { float t = sm[tid]; t = waveSum(t); if (tid == 0) bc = t; }
  __syncthreads();
  float inv = 1.f / bc;
  for (int i = tid; i < S; i += 1024) {
    float w = expf(scores[i] - M) * inv;
    w_ws[i] = w; w_out[i] = w;
  }
}

// K7: cpart[c][i] = sum_{r in chunk c} w[r]*enc[r][i]   (16 chunks of 512 rows)
__global__ void k_cpart(const float* __restrict__ enc, const float* __restrict__ w,
                        float* __restrict__ cpart) {
  int i = blockIdx.x * blockDim.x + threadIdx.x;
  if (i >= H) return;
  int r0 = blockIdx.y * 512;
  float a = 0.f;
  for (int r = r0; r < r0 + 512; ++r) a += w[r] * enc[(size_t)r * H + i];
  cpart[blockIdx.y * H + i] = a;
}
__global__ void k_creduce(const float* __restrict__ cpart, float* __restrict__ ctx_out,
                          float* __restrict__ joined) {
  int i = blockIdx.x * blockDim.x + threadIdx.x;
  if (i >= H) return;
  float a = 0.f;
  for (int c = 0; c < 16; ++c) a += cpart[c * H + i];
  ctx_out[i] = a; joined[H + i] = a;
}

// K8: vocab projection via WMMA f32 16x16x4. One wave = 16 vocab rows.
// A layout (16x4 f32): lanes 0-15 hold rows, K = k+2*khalf {+0,+1} in v[0],v[1].
// B replicated so every N column equals the joined-vector chunk -> D column = 16 dots.
__global__ __launch_bounds__(256) void k_wout(
    const float* __restrict__ Wout, const float* __restrict__ bout,
    const float* __restrict__ joined, float* __restrict__ logits) {
  const int lane  = threadIdx.x & 31;
  const int widx  = threadIdx.x >> 5;
  const int row_base = (blockIdx.x * 8 + widx) * 16;
  const int m     = lane & 15;
  const int khalf = lane >> 4;            // 0: K+0,1   1: K+2,3
  int row = row_base + m; if (row > V - 1) row = V - 1;   // branchless clamp, EXEC stays full
  const float* wrow = Wout + (size_t)row * IN + 2 * khalf;
  const float* bvec = joined + 2 * khalf;
  v8f c0 = {0.f,0.f,0.f,0.f,0.f,0.f,0.f,0.f};
  v8f c1 = {0.f,0.f,0.f,0.f,0.f,0.f,0.f,0.f};
  for (int k = 0; k < IN; k += 8) {
    v2f a0 = { wrow[k],     wrow[k + 1] };
    v2f b0 = { bvec[k],     bvec[k + 1] };
    v2f a1 = { wrow[k + 4], wrow[k + 5] };
    v2f b1 = { bvec[k + 4], bvec[k + 5] };
    c0 = __builtin_amdgcn_wmma_f32_16x16x4_f32(false, a0, false, b0, (short)0, c0, false, false);
    c1 = __builtin_amdgcn_wmma_f32_16x16x4_f32(false, a1, false, b1, (short)0, c1, false, false);
  }
  // D (16x16 f32): lane 0 -> M=0..7 in c[0..7]; lane 16 -> M=8..15. N columns identical.
  if ((lane & 15) == 0) {
    int base = row_base + (khalf ? 8 : 0);
    for (int g = 0; g < 8; ++g) {
      int r = base + g;
      if (r < V) logits[r] = c0[g] + c1[g] + bout[r];
    }
  }
}

// K9: log-sum-exp stats over 50257 (single block)
__global__ __launch_bounds__(1024) void k_lse(const float* __restrict__ logits,
                                              float* __restrict__ red) {
  __shared__ float sm[32]; __shared__ float bc;
  int tid = threadIdx.x;
  float m = -3.4e38f;
  for (int i = tid; i < V; i += 1024) m = fmaxf(m, logits[i]);
  m = waveMax(m);
  if ((tid & 31) == 0) sm[tid >> 5] = m;
  __syncthreads();
  if (tid < 32) { float t = sm[tid]; t = waveMax(t); if (tid == 0) bc = t; }
  __syncthreads();
  float M = bc;
  __syncthreads();
  float s = 0.f;
  for (int i = tid; i < V; i += 1024) s += expf(logits[i] - M);
  s = waveSum(s);
  if ((tid & 31) == 0) sm[tid >> 5] = s;
  __syncthreads();
  if (tid < 32) { float t = sm[tid]; t = waveSum(t); if (tid == 0) bc = t; }
  __syncthreads();
  if (tid == 0) { red[0] = M; red[1] = logf(bc); }
}

// K10: output[i] = logits[i] - M - log(sumexp)
__global__ void k_final(const float* __restrict__ logits, const float* __restrict__ red,
                        float* __restrict__ out) {
  int i = blockIdx.x * blockDim.x + threadIdx.x;
  if (i >= V) return;
  out[i] = logits[i] - red[0] - red[1];
}

// ---------------------------------------------------------------------------
extern "C" void kernel_launch(void* const* d_in, const int* in_sizes, int n_in,
                              void* d_out, int out_size, void* d_ws, size_t ws_size,
                              hipStream_t stream) {
  const int*   word = (const int*)  d_in[0];
  const float* lc   = (const float*)d_in[1];
  const float* h    = (const float*)d_in[2];
  const float* enc  = (const float*)d_in[3];
  const float* emb  = (const float*)d_in[4];
  const float* Wih  = (const float*)d_in[5];
  const float* Whh  = (const float*)d_in[6];
  const float* bih  = (const float*)d_in[7];
  const float* bhh  = (const float*)d_in[8];
  const float* Wa   = (const float*)d_in[9];
  const float* ba   = (const float*)d_in[10];
  const float* Wout = (const float*)d_in[11];
  const float* bout = (const float*)d_in[12];

  float* ws  = (float*)d_ws;
  float* out = (float*)d_out;

  // 1. embed + concat
  k_embed<<<(IN + 255) / 256, 256, 0, stream>>>(word, emb, lc, ws + WS_X);
  // 2. GRU gate GEMVs (3096 waves, 8 per block)
  k_gates<<<G3 / 8, 256, 0, stream>>>(Wih, Whh, bih, bhh, ws + WS_X, h,
                                      ws + WS_GX, ws + WS_GH);
  // 3. GRU elementwise -> h_new (+ d_out hidden, joined[0:H])
  k_gru<<<(H + 255) / 256, 256, 0, stream>>>(ws + WS_GX, ws + WS_GH, h,
                                             ws + WS_HNEW, ws + WS_JOIN, out + OUT_HID);
  // 4. v = W_a^T h_new (partials over 8 j-chunks), sb = b_a . h_new
  {
    dim3 g((H + 255) / 256, 8);
    k_vpart<<<g, 256, 0, stream>>>(Wa, ws + WS_HNEW, ws + WS_VPART);
  }
  k_vreduce<<<(H + 255) / 256, 256, 0, stream>>>(ws + WS_VPART, ws + WS_V);
  k_sb<<<1, 1024, 0, stream>>>(ba, ws + WS_HNEW, ws + WS_SB);
  // 5. scores = enc . v + sb (8192 waves)
  k_scores<<<S / 8, 256, 0, stream>>>(enc, ws + WS_V, ws + WS_SB, ws + WS_SCORES);
  // 6. softmax -> weights (ws + d_out attn_w)
  k_softmax<<<1, 1024, 0, stream>>>(ws + WS_SCORES, ws + WS_ATTNW, out + OUT_ATTNW);
  // 7. attn_context = w^T enc (partials over 16 row-chunks) -> d_out ctx, joined[H:2H]
  {
    dim3 g((H + 255) / 256, 16);
    k_cpart<<<g, 256, 0, stream>>>(enc, ws + WS_ATTNW, ws + WS_CPART);
  }
  k_creduce<<<(H + 255) / 256, 256, 0, stream>>>(ws + WS_CPART, out + OUT_CTX, ws + WS_JOIN);
  // 8. vocab projection via WMMA (415 MB stream, the bandwidth-dominant stage)
  k_wout<<<(V + 127) / 128, 256, 0, stream>>>(Wout, bout, ws + WS_JOIN, ws + WS_LOGITS);
  // 9-10. log_softmax
  k_lse<<<1, 1024, 0, stream>>>(ws + WS_LOGITS, ws + WS_RED);
  k_final<<<(V + 255) / 256, 256, 0, stream>>>(ws + WS_LOGITS, ws + WS_RED, out + OUT_OUTPUT);
}